// LatentCircuitModel_40518721470624
// MI455X (gfx1250) — compile-verified
//
#include <hip/hip_runtime.h>
#include <math.h>

#define N_LAT 16
#define N_OBS 1024
#define BATCH 32
#define SEQLEN 2048

typedef float v2f __attribute__((ext_vector_type(2)));
typedef float v8f __attribute__((ext_vector_type(8)));

// Native CDNA5 transcendental tanh (TRANS op, single instruction).
__device__ __forceinline__ float fast_tanh(float x) {
#if defined(__gfx1250__)
    float y;
    asm("v_tanh_f32 %0, %1" : "=v"(y) : "v"(x));
    return y;
#elif __has_builtin(__builtin_amdgcn_tanhf)
    return __builtin_amdgcn_tanhf(x);
#else
    return tanhf(x);
#endif
}

// D(16x16) += A(16x16)*B(16x16) accumulated as 4 chained K=4 f32 WMMAs.
__device__ __forceinline__ v8f wmma_chain_k16(const v2f a[4], const v2f b[4], v8f c) {
#pragma unroll
    for (int j = 0; j < 4; ++j) {
        c = __builtin_amdgcn_wmma_f32_16x16x4_f32(
            /*neg_a=*/false, a[j], /*neg_b=*/false, b[j],
            /*c_mod=*/(short)0, c, /*reuse_a=*/false, /*reuse_b=*/false);
    }
    return c;
}

// ---------------------------------------------------------------------------
// Kernel 1: drive[bt,k] = sum_l x[bt,l] * W_in[k,l] + b_rec[k]
// One wave per 16-row bt tile; 4096 tiles / 8 waves per block = 512 blocks.
// ---------------------------------------------------------------------------
__global__ __launch_bounds__(256) void drive_kernel(const float* __restrict__ x,
                                                    const float* __restrict__ W_in,
                                                    const float* __restrict__ b_rec,
                                                    float* __restrict__ drive) {
    const int wave   = threadIdx.x >> 5;
    const int lane   = threadIdx.x & 31;
    const int laneLo = lane & 15;
    const int laneHi = lane >> 4;
    const int tile   = blockIdx.x * 8 + wave;

    const float* xrow = x + (size_t)(tile * 16 + laneLo) * N_LAT;
    const float* wrow = W_in + laneLo * N_LAT;

    v2f a[4], b[4];
#pragma unroll
    for (int j = 0; j < 4; ++j) {
        const int k = 4 * j + 2 * laneHi;       // ISA 32-bit A layout
        a[j].x = xrow[k];
        a[j].y = xrow[k + 1];
        b[j].x = wrow[k];
        b[j].y = wrow[k + 1];
    }

    v8f c = {};
    c = wmma_chain_k16(a, b, c);

    const float bias = b_rec[laneLo];
    float* dst = drive + (size_t)(tile * 16) * N_LAT + laneLo;
#pragma unroll
    for (int v = 0; v < 8; ++v) {
        const int m = v + 8 * laneHi;           // ISA C/D layout
        dst[m * N_LAT] = c[v] + bias;
    }
}

// ---------------------------------------------------------------------------
// Kernel 2: sequential tanh recurrence.  lane = batch (wave32 == 32 batches).
// W_eff^T in VGPRs; next-step drive software-pipelined; native v_tanh_f32.
// ---------------------------------------------------------------------------
__global__ __launch_bounds__(32, 1) void scan_kernel(const float* __restrict__ drive,
                                                     const float* __restrict__ W_rec,
                                                     const float* __restrict__ ei,
                                                     float* __restrict__ hidden) {
    const int b = threadIdx.x;

    // W[l][k] = |W_rec[k][l]| * ei[k][l]  -- loop-invariant, register-resident.
    float W[N_LAT][N_LAT];
#pragma unroll
    for (int l = 0; l < N_LAT; ++l) {
#pragma unroll
        for (int k = 0; k < N_LAT; ++k) {
            W[l][k] = fabsf(W_rec[k * N_LAT + l]) * ei[k * N_LAT + l];
        }
    }

    float h[N_LAT];
#pragma unroll
    for (int k = 0; k < N_LAT; ++k) h[k] = 0.0f;

    const float4* dp = (const float4*)(drive + (size_t)b * SEQLEN * N_LAT);
    float4* hp = (float4*)(hidden + (size_t)b * SEQLEN * N_LAT);

    float4 u0 = dp[0], u1 = dp[1], u2 = dp[2], u3 = dp[3];

    // one recurrence step: h = tanh(W_eff @ h + u); writes h to *hp
    auto step = [&](const float4& s0, const float4& s1,
                    const float4& s2, const float4& s3) {
        float acc[N_LAT] = {s0.x, s0.y, s0.z, s0.w,
                            s1.x, s1.y, s1.z, s1.w,
                            s2.x, s2.y, s2.z, s2.w,
                            s3.x, s3.y, s3.z, s3.w};
#pragma unroll
        for (int l = 0; l < N_LAT; ++l) {
            const float hl = h[l];
#pragma unroll
            for (int k = 0; k < N_LAT; ++k) acc[k] += W[l][k] * hl;
        }
#pragma unroll
        for (int k = 0; k < N_LAT; ++k) h[k] = fast_tanh(acc[k]);

        hp[0] = make_float4(h[0],  h[1],  h[2],  h[3]);
        hp[1] = make_float4(h[4],  h[5],  h[6],  h[7]);
        hp[2] = make_float4(h[8],  h[9],  h[10], h[11]);
        hp[3] = make_float4(h[12], h[13], h[14], h[15]);
        hp += 4;
    };

    for (int t = 0; t < SEQLEN - 1; ++t) {
        // speculative prefetch ~8 steps ahead (drive is L2-resident)
        __builtin_prefetch((const void*)(dp + 32), 0, 0);

        const float4 n0 = dp[4], n1 = dp[5], n2 = dp[6], n3 = dp[7];
        dp += 4;

        step(u0, u1, u2, u3);

        u0 = n0; u1 = n1; u2 = n2; u3 = n3;
    }
    step(u0, u1, u2, u3);   // final step, no further load
}

// ---------------------------------------------------------------------------
// Kernel 3: outputs[bt,o] = sum_l hidden[bt,l] * W_out[o,l] + b_out[o]
// Block (8 waves) owns one bt-tile; each wave does 8 o-tiles reusing A.
// Pure store-bandwidth bound: non-temporal stores keep L2 for W_out/hidden.
// ---------------------------------------------------------------------------
__global__ __launch_bounds__(256) void out_kernel(const float* __restrict__ hidden,
                                                  const float* __restrict__ W_out,
                                                  const float* __restrict__ b_out,
                                                  float* __restrict__ out) {
    const int wave   = threadIdx.x >> 5;
    const int lane   = threadIdx.x & 31;
    const int laneLo = lane & 15;
    const int laneHi = lane >> 4;
    const int btTile = blockIdx.x;

    const float* hrow = hidden + (size_t)(btTile * 16 + laneLo) * N_LAT;
    v2f a[4];
#pragma unroll
    for (int j = 0; j < 4; ++j) {
        const int k = 4 * j + 2 * laneHi;
        a[j].x = hrow[k];
        a[j].y = hrow[k + 1];
    }

#pragma unroll 1
    for (int i = 0; i < 8; ++i) {
        const int otile = wave * 8 + i;
        const int obase = otile * 16;
        const float* wrow = W_out + (size_t)(obase + laneLo) * N_LAT;

        v2f bb[4];
#pragma unroll
        for (int j = 0; j < 4; ++j) {
            const int k = 4 * j + 2 * laneHi;
            bb[j].x = wrow[k];
            bb[j].y = wrow[k + 1];
        }

        v8f c = {};
        c = wmma_chain_k16(a, bb, c);

        const float bias = b_out[obase + laneLo];
        float* dst = out + (size_t)(btTile * 16) * N_OBS + obase + laneLo;
#pragma unroll
        for (int v = 0; v < 8; ++v) {
            const int m = v + 8 * laneHi;
            __builtin_nontemporal_store(c[v] + bias, &dst[(size_t)m * N_OBS]);
        }
    }
}

// ---------------------------------------------------------------------------
extern "C" void kernel_launch(void* const* d_in, const int* in_sizes, int n_in,
                              void* d_out, int out_size, void* d_ws, size_t ws_size,
                              hipStream_t stream) {
    (void)in_sizes; (void)n_in; (void)out_size; (void)ws_size;

    const float* x     = (const float*)d_in[0];  // (32, 2048, 16)
    const float* W_rec = (const float*)d_in[1];  // (16, 16)
    const float* W_in  = (const float*)d_in[2];  // (16, 16)
    const float* W_out = (const float*)d_in[3];  // (1024, 16)
    const float* b_rec = (const float*)d_in[4];  // (16,)
    const float* b_out = (const float*)d_in[5];  // (1024,)
    const float* ei    = (const float*)d_in[6];  // (16, 16)

    float* outputs = (float*)d_out;                               // (32,2048,1024)
    float* hidden  = outputs + (size_t)BATCH * SEQLEN * N_OBS;    // (32,2048,16)
    float* drive   = (float*)d_ws;                                // (32,2048,16)

    drive_kernel<<<512, 256, 0, stream>>>(x, W_in, b_rec, drive);
    scan_kernel<<<1, 32, 0, stream>>>(drive, W_rec, ei, hidden);
    out_kernel<<<4096, 256, 0, stream>>>(hidden, W_out, b_out, outputs);
}